// FCDynamic_27144193311414
// MI455X (gfx1250) — compile-verified
//
#include <hip/hip_runtime.h>
#include <stdint.h>

// ---------------------------------------------------------------------------
// Hash-join expand for MI455X (gfx1250).
//
// Pure data-movement workload: ~32 MB fact stream in, 128 MB padded expansion
// out -> ~7 us roofline at 23.3 TB/s HBM. Zero matrix FLOPs, so WMMA does not
// apply; the CDNA5-specific paths exercised are the Tensor Data Mover
// (tensor_load_to_lds, one descriptor per 256-fact tile, TENSORcnt wait) and
// L2-resident atomic top-k insertion.
//
// Exactness: "first k_max facts of a bucket in original order" == "k_max
// smallest fact indices of that bucket, ascending". The concurrent atomicMin
// insertion cascade computes exactly that, independent of interleaving, so
// the result is deterministic and bit-matches the reference's stable sort.
// ---------------------------------------------------------------------------

#define EMAX 4096               // max distinct subjects supported (E = 2000)
#define SLOT_EMPTY 0xFFFFFFFFu

typedef unsigned u32x4 __attribute__((ext_vector_type(4)));
typedef unsigned u32x8 __attribute__((ext_vector_type(8)));

__global__ void k_init_slots(unsigned* __restrict__ slots, int n) {
  int t = blockIdx.x * blockDim.x + threadIdx.x;
  if (t < n) slots[t] = SLOT_EMPTY;
}

// One TDM tile copy per block: facts[block_start .. block_start+255] -> LDS.
__global__ __launch_bounds__(256)
void k_insert(const int* __restrict__ facts,        // [F,3] (pred,subj,obj)
              const int* __restrict__ pred_idx_p,   // scalar
              unsigned* __restrict__ slots,         // [EMAX,K]
              int F, int K) {
  __shared__ int lds_facts[256 * 3];

  const int block_start = blockIdx.x * 256;
  const int nfacts = min(256, F - block_start);
  const unsigned tile_elems = (unsigned)(nfacts * 3);     // dwords to move

  if (threadIdx.x == 0) {
    // Build Tensor DMA Descriptor (D#), groups 0 and 1 (1-D tile, 4B elems).
    const unsigned long long gaddr =
        (unsigned long long)(uintptr_t)facts +
        (unsigned long long)block_start * 12ull;
    const unsigned lds_base = (unsigned)(size_t)(&lds_facts[0]);

    u32x4 g0;
    g0[0] = 1u;                                           // count=1, user mode
    g0[1] = lds_base;                                     // lds_addr
    g0[2] = (unsigned)(gaddr & 0xFFFFFFFFull);            // global_addr[31:0]
    g0[3] = (unsigned)((gaddr >> 32) & 0x01FFFFFFull)     // global_addr[56:32]
            | 0x80000000u;                                // type=2 ("image")

    u32x8 g1;
    g1[0] = (2u << 16);                                   // wg_mask=0, 4B elems
    g1[1] = (tile_elems & 0xFFFFu) << 16;                 // tensor_dim0[15:0]
    g1[2] = (tile_elems >> 16) | (1u << 16);              // dim0 hi | dim1=1
    g1[3] = (tile_elems & 0xFFFFu) << 16;                 // dim1 hi=0|tile_dim0
    g1[4] = 0u;                                           // tile_dim1/2 unused
    g1[5] = tile_elems;                                   // dim0_stride lo
    g1[6] = 0u;                                           // stride hi / dim1
    g1[7] = 0u;

    asm volatile("tensor_load_to_lds %0, %1" :: "s"(g0), "s"(g1) : "memory");
    __builtin_amdgcn_s_wait_tensorcnt(0);                 // TENSORcnt == 0
  }
  __syncthreads();                                        // LDS tile visible

  const int i = block_start + (int)threadIdx.x;
  if (i >= F) return;
  const int p = pred_idx_p[0];
  const int pred = lds_facts[threadIdx.x * 3 + 0];
  if (pred != p) return;                                  // ~1/P survive
  const int s = lds_facts[threadIdx.x * 3 + 1];
  if (s < 0 || s >= EMAX) return;

  // Bounded atomicMin insertion cascade: keeps K smallest indices, sorted.
  unsigned cur = (unsigned)i;
  unsigned* row = slots + (size_t)s * (size_t)K;
  for (int j = 0; j < K; ++j) {
    unsigned old = atomicMin(&row[j], cur);
    if (old == SLOT_EMPTY) break;                         // landed in empty slot
    cur = (old > cur) ? old : cur;                        // carry displaced idx
  }
}

// Materialize the per-subject result table once (the only scattered gathers).
__global__ void k_build_table(const int* __restrict__ facts,
                              const float* __restrict__ fw,
                              const unsigned* __restrict__ slots,
                              int* __restrict__ res_obj,
                              float* __restrict__ res_w, int n) {
  int t = blockIdx.x * blockDim.x + threadIdx.x;
  if (t >= n) return;
  const unsigned idx = slots[t];
  int obj = -1;
  float w = 0.0f;
  if (idx != SLOT_EMPTY) {
    obj = facts[(size_t)idx * 3 + 2];
    w = fw[idx];
  }
  res_obj[t] = obj;
  res_w[t] = w;
}

__device__ __forceinline__ int subject_of(int v, int p, int E, int P) {
  // key = clip(p*E + v, 0, P*E-1); recover subject within pred row.
  long long key = (long long)p * E + (long long)v;
  const long long hi = (long long)P * E - 1;
  key = (key < 0) ? 0 : (key > hi ? hi : key);
  long long s = key - (long long)p * E;
  s = (s < 0) ? 0 : (s >= E ? (long long)E - 1 : s);
  return (int)s;
}

// Vectorized broadcast: each thread moves one int4 + one float4 row chunk.
__global__ __launch_bounds__(256)
void k_expand_v4(const int* __restrict__ key_vals,
                 const int* __restrict__ pred_idx_p,
                 const int* __restrict__ Ep, const int* __restrict__ Pp,
                 const int* __restrict__ res_obj,
                 const float* __restrict__ res_w,
                 int* __restrict__ obj_out, float* __restrict__ w_out,
                 int N, int K) {
  const int KV = K >> 2;                                  // int4 chunks / query
  const long long t = (long long)blockIdx.x * blockDim.x + threadIdx.x;
  const long long total = (long long)N * (long long)KV;
  if (t >= total) return;
  const int q = (int)(t / KV);
  const int c = (int)(t - (long long)q * KV);

  const int s = subject_of(key_vals[q], pred_idx_p[0], Ep[0], Pp[0]);

  int4 o = make_int4(-1, -1, -1, -1);
  float4 wv = make_float4(0.0f, 0.0f, 0.0f, 0.0f);
  if (s < EMAX) {
    const size_t src = (size_t)s * (size_t)K + (size_t)c * 4;
    o  = *(const int4*)(res_obj + src);                   // L2-hot 2MB table
    wv = *(const float4*)(res_w + src);
  }
  const size_t dst = (size_t)q * (size_t)K + (size_t)c * 4;
  *(int4*)(obj_out + dst) = o;                            // b128 coalesced
  *(float4*)(w_out + dst) = wv;
}

// Scalar fallback for K not divisible by 4.
__global__ __launch_bounds__(256)
void k_expand_scalar(const int* __restrict__ key_vals,
                     const int* __restrict__ pred_idx_p,
                     const int* __restrict__ Ep, const int* __restrict__ Pp,
                     const int* __restrict__ res_obj,
                     const float* __restrict__ res_w,
                     int* __restrict__ obj_out, float* __restrict__ w_out,
                     int N, int K) {
  const long long t = (long long)blockIdx.x * blockDim.x + threadIdx.x;
  const long long total = (long long)N * (long long)K;
  if (t >= total) return;
  const int q = (int)(t / K);
  const int k = (int)(t - (long long)q * K);
  const int s = subject_of(key_vals[q], pred_idx_p[0], Ep[0], Pp[0]);
  int obj = -1;
  float w = 0.0f;
  if (s < EMAX) {
    obj = res_obj[(size_t)s * K + k];
    w = res_w[(size_t)s * K + k];
  }
  obj_out[t] = obj;
  w_out[t] = w;
}

extern "C" void kernel_launch(void* const* d_in, const int* in_sizes, int n_in,
                              void* d_out, int out_size, void* d_ws, size_t ws_size,
                              hipStream_t stream) {
  const int*   facts    = (const int*)  d_in[0];   // facts_idx [F,3] int32
  const float* fw       = (const float*)d_in[1];   // fact_w    [F]   f32
  const int*   key_vals = (const int*)  d_in[2];   // key_vals  [N]   int32
  const int*   pred_idx = (const int*)  d_in[3];   // scalar
  const int*   Ep       = (const int*)  d_in[4];   // scalar
  const int*   Pp       = (const int*)  d_in[5];   // scalar
  // d_in[6] = k_max; K also follows from out_size = N*K*2.

  const int F = in_sizes[0] / 3;
  const int N = in_sizes[2];
  const int K = out_size / (2 * N);

  // Workspace: slots | res_obj | res_w  (each EMAX*K dwords; 1.5 MB @ K=32)
  const int nslots = EMAX * K;
  unsigned* slots   = (unsigned*)d_ws;
  int*      res_obj = (int*)(slots + nslots);
  float*    res_w   = (float*)(res_obj + nslots);

  int*   obj_out = (int*)d_out;
  float* w_out   = (float*)((char*)d_out + (size_t)N * (size_t)K * sizeof(int));

  k_init_slots<<<(nslots + 255) / 256, 256, 0, stream>>>(slots, nslots);
  k_insert<<<(F + 255) / 256, 256, 0, stream>>>(facts, pred_idx, slots, F, K);
  k_build_table<<<(nslots + 255) / 256, 256, 0, stream>>>(facts, fw, slots,
                                                          res_obj, res_w, nslots);

  if ((K & 3) == 0) {
    const long long total = (long long)N * (long long)(K >> 2);
    const int blocks = (int)((total + 255) / 256);
    k_expand_v4<<<blocks, 256, 0, stream>>>(key_vals, pred_idx, Ep, Pp,
                                            res_obj, res_w, obj_out, w_out, N, K);
  } else {
    const long long total = (long long)N * (long long)K;
    const int blocks = (int)((total + 255) / 256);
    k_expand_scalar<<<blocks, 256, 0, stream>>>(key_vals, pred_idx, Ep, Pp,
                                                res_obj, res_w, obj_out, w_out, N, K);
  }
}